// CTRNNpad_58445914964034
// MI455X (gfx1250) — compile-verified
//
#include <hip/hip_runtime.h>

// CTRNN forward on gfx1250 (MI455X).
// B=256, T=2048, D=32, H=64.  seq_lengths only gates gradients -> ignored
// (forward value identical, per the reference comment).
// d_out = outputs[B,T,H] ++ h_last[B,H], float32.
//
// Latency-bound on the T=2048 dependent scan (FLOPs ~6.4 G, bytes ~200 MB =
// ~8.6 us at 23.3 TB/s), so everything targets the per-step critical path:
//   grid = 16 blocks (one per 16-batch-row tile), block = 128 = 4 wave32s,
//   wave w owns h columns [16w,16w+16) with v_wmma_f32_16x16x4_f32 (exact fp32).
//   - Weight B-operand slices live in VGPRs for the whole kernel.
//   - h is exchanged per step via double-buffered LDS; C-layout -> A-layout
//     transpose is exactly one ds_load_2addr_b64 per 2 K-chunks.
//   - A sched_barrier pins all 8 LDS loads into one clause BEFORE the WMMA
//     burst (one LDS round-trip on the critical path, not seven).
//   - THREE accumulator chains: x-projection (off critical path, absorbs DS
//     latency) + two independent 8-deep h chains the scheduler can interleave.
//   - x loads + bias fold run off the critical path.

typedef __attribute__((ext_vector_type(2))) float v2f;
typedef __attribute__((ext_vector_type(8))) float v8f;

constexpr int B = 256;
constexpr int T = 2048;
constexpr int D = 32;
constexpr int H = 64;
constexpr float ALPHA = 16.67f / 40.0f;
constexpr float OMA   = 1.0f - ALPHA;

__global__ __launch_bounds__(128) void ctrnn_scan_kernel(
    const float* __restrict__ x,     // [B, T, D]
    const float* __restrict__ W_in,  // [H, D]
    const float* __restrict__ b_in,  // [H]
    const float* __restrict__ W_hh,  // [H, H]
    const float* __restrict__ b_hh,  // [H]
    float* __restrict__ out)         // [B*T*H] outputs ++ [B*H] h_last
{
    // Double-buffered h state for this 16-row batch tile: [2][16 rows][H cols]
    __shared__ float hbuf[2][16 * H];

    const int lane = threadIdx.x & 31;
    const int wave = threadIdx.x >> 5;   // N-tile index 0..3 (h cols 16w..16w+15)
    const int l16  = lane & 15;
    const int half = lane >> 4;          // 0: lanes 0-15, 1: lanes 16-31
    const int n    = wave * 16 + l16;    // h column handled by this lane (B/C layout)
    const int row  = l16;                // batch row within tile (A layout)
    const int b0   = blockIdx.x * 16;    // first batch row of this tile

    // ---- Preload B operands (weights) into VGPRs, once. ----
    // B layout for 16x16x4 f32: V0 = {K=4kc | 4kc+2}, V1 = {K+1 | K+3}; per lane
    // the pair {W^T[K][n], W^T[K+1][n]} = {W[n][K], W[n][K+1]} is a contiguous
    // 8-byte-aligned float2 in the row-major weight matrix.
    v2f bwin[D / 4];   // W_in^T slices, K = 0..31
    v2f bwhh[H / 4];   // W_hh^T slices, K = 0..63
#pragma unroll
    for (int kc = 0; kc < D / 4; ++kc)
        bwin[kc] = *reinterpret_cast<const v2f*>(&W_in[n * D + 4 * kc + 2 * half]);
#pragma unroll
    for (int kc = 0; kc < H / 4; ++kc)
        bwhh[kc] = *reinterpret_cast<const v2f*>(&W_hh[n * H + 4 * kc + 2 * half]);

    const float bias = b_in[n] + b_hh[n];

    // ---- Zero initial h state (h0 = 0). ----
    for (int i = threadIdx.x; i < 16 * H; i += 128) hbuf[0][i] = 0.0f;

    v8f hC;                    // current h in C layout: (M = v + 8*half, N = n)
#pragma unroll
    for (int v = 0; v < 8; ++v) hC[v] = 0.0f;

    const float* xrow = x + (size_t)(b0 + row) * T * D;  // this lane's A-row base
    float* hlast = out + (size_t)B * T * H;

    for (int t = 0; t < T; ++t) {
        const int p = t & 1;

        // x A-operand loads: independent of h -> issued first so the VMEM
        // latency overlaps the h exchange + other waves' barrier arrival.
        v2f ax[D / 4];
#pragma unroll
        for (int kc = 0; kc < D / 4; ++kc)
            ax[kc] = *reinterpret_cast<const v2f*>(&xrow[t * D + 4 * kc + 2 * half]);

        // Chain X: accX = bias + x_t @ W_in^T  (bias folded into C init; this
        // chain runs in the shadow of the barrier / DS-load returns).
        v8f accX;
#pragma unroll
        for (int v = 0; v < 8; ++v) accX[v] = bias;
#pragma unroll
        for (int kc = 0; kc < D / 4; ++kc)
            accX = __builtin_amdgcn_wmma_f32_16x16x4_f32(
                false, ax[kc], false, bwin[kc], (short)0, accX, false, false);

        // hbuf[p] now holds h_t from all 4 waves (or the zero init at t==0).
        __syncthreads();

        // Stage ALL 16 h A-operand pairs into distinct registers: one clause
        // of 8 ds_load_2addr_b64.  The sched_barrier stops the post-RA
        // scheduler from sinking these loads back between the WMMAs (which
        // would serialize LDS latency onto the critical path 7 extra times).
        v2f ah[H / 4];
#pragma unroll
        for (int kc = 0; kc < H / 4; ++kc)
            ah[kc] = *reinterpret_cast<const v2f*>(
                &hbuf[p][row * H + 4 * kc + 2 * half]);
        __builtin_amdgcn_sched_barrier(0);  // nothing crosses: loads stay above

        // Chains A/B: h_t @ W_hh^T split into two INDEPENDENT 8-deep
        // accumulation chains (even/odd K-chunks), both starting from the
        // inline-0 C operand, so the scheduler can interleave them and hide
        // the dependent-accumulation latency.
        v8f accA, accB;
#pragma unroll
        for (int v = 0; v < 8; ++v) { accA[v] = 0.0f; accB[v] = 0.0f; }
#pragma unroll
        for (int kc = 0; kc < H / 4; kc += 2) {
            accA = __builtin_amdgcn_wmma_f32_16x16x4_f32(
                false, ah[kc],     false, bwhh[kc],     (short)0, accA, false, false);
            accB = __builtin_amdgcn_wmma_f32_16x16x4_f32(
                false, ah[kc + 1], false, bwhh[kc + 1], (short)0, accB, false, false);
        }

        // h_{t+1} = relu(h_t*(1-alpha) + (accX+accA+accB)*alpha);
        // the mask in the reference is a forward no-op.
#pragma unroll
        for (int v = 0; v < 8; ++v)
            hC[v] = fmaxf(fmaf(accX[v] + accA[v] + accB[v], ALPHA, hC[v] * OMA),
                          0.0f);

        // Publish h_{t+1} into the other LDS buffer and stream outputs[b,t,:].
        // (Stores to buffer p^1 cannot collide with waves still reading p; the
        //  single barrier per step keeps all waves within one phase.)
#pragma unroll
        for (int v = 0; v < 8; ++v) {
            const int m = v + 8 * half;                      // batch row in tile
            hbuf[p ^ 1][m * H + n] = hC[v];
            out[((size_t)(b0 + m) * T + t) * H + n] = hC[v]; // coalesced per half-wave
        }
    }

    // h_last
#pragma unroll
    for (int v = 0; v < 8; ++v) {
        const int m = v + 8 * half;
        hlast[(size_t)(b0 + m) * H + n] = hC[v];
    }
}

extern "C" void kernel_launch(void* const* d_in, const int* in_sizes, int n_in,
                              void* d_out, int out_size, void* d_ws, size_t ws_size,
                              hipStream_t stream) {
    const float* x    = (const float*)d_in[0];
    // d_in[1] = seq_lengths (int32): forward output is mask-independent -> unused.
    const float* W_in = (const float*)d_in[2];
    const float* b_in = (const float*)d_in[3];
    const float* W_hh = (const float*)d_in[4];
    const float* b_hh = (const float*)d_in[5];
    float* out = (float*)d_out;

    ctrnn_scan_kernel<<<dim3(B / 16), dim3(128), 0, stream>>>(
        x, W_in, b_in, W_hh, b_hh, out);
}